// GDMLTorchPredict_86784109183204
// MI455X (gfx1250) — compile-verified
//
#include <hip/hip_runtime.h>
#include <math.h>

// ---------------------------------------------------------------------------
// GDML predict: Es(128), Fs(128,21,3).  fp32 WMMA (V_WMMA_F32_16X16X4_F32).
// ---------------------------------------------------------------------------

typedef __attribute__((ext_vector_type(2))) float v2f;
typedef __attribute__((ext_vector_type(8))) float v8f;

constexpr int Mn = 128;    // molecules
constexpr int NA = 21;     // atoms
constexpr int Dn = 210;    // descriptors = 21*20/2
constexpr int Tn = 4096;   // training points
constexpr int KP = 224;    // Dn padded (multiple of 16, zero-filled)
constexpr int DP = 224;    // output-D padding for GEMM2 tiles

constexpr float Qf   = 0.22360679774997896f;          // sqrt(5)/SIG, SIG=10
constexpr float PREF = 5.0f / (3.0f * 10.0f * 10.0f); // 5/(3*sig^2)
constexpr float STDf = 1.0f;
constexpr float Cf   = 0.0f;

// ---- workspace layout (floats), each region 64-float aligned --------------
constexpr size_t rup(size_t n) { return (n + 63) & ~(size_t)63; }
constexpr size_t OFF_QXSA = 0;                                   // M x KP
constexpr size_t OFF_R    = OFF_QXSA + rup((size_t)Mn * KP);     // M x Dn
constexpr size_t OFF_XN   = OFF_R    + rup((size_t)Mn * Dn);     // M
constexpr size_t OFF_TN   = OFF_XN   + rup(Mn);                  // T
constexpr size_t OFF_TD   = OFF_TN   + rup(Tn);                  // T
constexpr size_t OFF_BT1  = OFF_TD   + rup(Tn);                  // KP x T (k-pairs)
constexpr size_t OFF_BJ1  = OFF_BT1  + rup((size_t)KP * Tn);     // KP x T
constexpr size_t OFF_BT2  = OFF_BJ1  + rup((size_t)KP * Tn);     // T x DP (t-pairs)
constexpr size_t OFF_BJ2  = OFF_BT2  + rup((size_t)Tn * DP);     // T x DP
constexpr size_t OFF_W1   = OFF_BJ2  + rup((size_t)Tn * DP);     // M x T
constexpr size_t OFF_W2   = OFF_W1   + rup((size_t)Mn * Tn);     // M x T
constexpr size_t OFF_E    = OFF_W2   + rup((size_t)Mn * Tn);     // M x T
constexpr size_t OFF_F1A  = OFF_E    + rup((size_t)Mn * Tn);     // M x DP
constexpr size_t OFF_F2   = OFF_F1A  + rup((size_t)Mn * DP);     // M x DP
constexpr size_t OFF_S1   = OFF_F2   + rup((size_t)Mn * DP);     // M
constexpr size_t OFF_ES   = OFF_S1   + rup(Mn);                  // M
constexpr size_t OFF_COEF = OFF_ES   + rup(Mn);                  // M x Dn

// ---- fp32 WMMA wrapper: D = A(16x4) * B(4x16) + C(16x16) ------------------
__device__ __forceinline__ v8f wmma_f32(v2f a, v2f b, v8f c) {
  // 8 args: (neg_a, A, neg_b, B, c_mod, C, reuse_a, reuse_b)
  return __builtin_amdgcn_wmma_f32_16x16x4_f32(false, a, false, b,
                                               (short)0, c, false, false);
}

// ---------------------------------------------------------------------------
// Kernel 1: per-molecule descriptors. qxs = q / r, padded to KP with zeros.
// ---------------------------------------------------------------------------
__global__ void prep_mol(const float* __restrict__ Rs, float* __restrict__ ws) {
  __shared__ float rs[NA * 3];
  __shared__ float red[256];
  const int m = blockIdx.x;
  const int tid = threadIdx.x;
  if (tid < NA * 3) rs[tid] = Rs[(size_t)m * NA * 3 + tid];
  __syncthreads();
  float part = 0.0f;
  if (tid < KP) {
    float qxs = 0.0f;
    if (tid < Dn) {
      // inverse of d = i*(i-1)/2 + j  (tril, k=-1, row-major)
      int i = (int)floorf(0.5f * (1.0f + sqrtf(1.0f + 8.0f * (float)tid)));
      while (i * (i - 1) / 2 > tid) --i;
      while ((i + 1) * i / 2 <= tid) ++i;
      int j = tid - i * (i - 1) / 2;
      float dx = rs[i * 3 + 0] - rs[j * 3 + 0];
      float dy = rs[i * 3 + 1] - rs[j * 3 + 1];
      float dz = rs[i * 3 + 2] - rs[j * 3 + 2];
      float r = sqrtf(dx * dx + dy * dy + dz * dz);
      ws[OFF_R + (size_t)m * Dn + tid] = r;
      qxs = Qf / r;
      part = qxs * qxs;
    }
    ws[OFF_QXSA + (size_t)m * KP + tid] = qxs;
  }
  red[tid] = part;
  __syncthreads();
  for (int s = 128; s > 0; s >>= 1) {
    if (tid < s) red[tid] += red[tid + s];
    __syncthreads();
  }
  if (tid == 0) ws[OFF_XN + m] = red[0];
}

// ---------------------------------------------------------------------------
// Kernel 2: per-training-point norms/dots + B operands packed as (k,k+1)
// float2 pairs in both orientations (single b64 load per WMMA B fragment).
// ---------------------------------------------------------------------------
__global__ void prep_train(const float* __restrict__ xs_train,
                           const float* __restrict__ Jx,
                           float* __restrict__ ws) {
  __shared__ float sN[256], sD[256];
  const int t = blockIdx.x;
  const int d = threadIdx.x;
  float pn = 0.0f, pd = 0.0f;
  if (d < KP) {
    float qxt = 0.0f, jx = 0.0f;
    if (d < Dn) {
      qxt = Qf * xs_train[(size_t)t * Dn + d];
      jx  = Jx[(size_t)t * Dn + d];
      pn = qxt * qxt;
      pd = qxt * jx;
    }
    // GEMM1 B (K over d): float2 pair index (d>>1)*T + t, component d&1
    size_t f1 = 2 * ((size_t)(d >> 1) * Tn + t) + (size_t)(d & 1);
    ws[OFF_BT1 + f1] = qxt;
    ws[OFF_BJ1 + f1] = jx;
    // GEMM2 B (K over t): float2 pair index (t>>1)*DP + d, component t&1
    size_t f2 = 2 * ((size_t)(t >> 1) * DP + d) + (size_t)(t & 1);
    ws[OFF_BT2 + f2] = qxt;
    ws[OFF_BJ2 + f2] = jx;
  }
  sN[d] = pn; sD[d] = pd;
  __syncthreads();
  for (int s = 128; s > 0; s >>= 1) {
    if (d < s) { sN[d] += sN[d + s]; sD[d] += sD[d + s]; }
    __syncthreads();
  }
  if (d == 0) { ws[OFF_TN + t] = sN[0]; ws[OFF_TD + t] = sD[0]; }
}

// ---------------------------------------------------------------------------
// Kernel 3: QX = qxs @ qxtT, QJ = qxs @ JxT  (both 128x4096, K=224),
// with fused epilogue: x_dist, exp, w1, w2, e — all in registers.
// One 16x16 tile per wave; 2048 tiles = 256 blocks x 8 waves.
// ---------------------------------------------------------------------------
__global__ void gemm1(float* __restrict__ ws) {
  const int wave = threadIdx.x >> 5;
  const int lane = threadIdx.x & 31;
  const int tile = blockIdx.x * 8 + wave;  // 0..2047
  const int mt = tile >> 8;                // 0..7
  const int tt = tile & 255;               // 0..255
  const int m0 = mt * 16, t0 = tt * 16;
  const int row = lane & 15, hi = lane >> 4;

  const float* A  = ws + OFF_QXSA + (size_t)(m0 + row) * KP + 2 * hi;
  const float* B1 = ws + OFF_BT1 + (size_t)2 * hi * Tn + 2 * (t0 + row);
  const float* B2 = ws + OFF_BJ1 + (size_t)2 * hi * Tn + 2 * (t0 + row);

  v8f cX = {};
  v8f cJ = {};
#pragma unroll 8
  for (int k = 0; k < KP; k += 4) {
    v2f a  = *(const v2f*)(A + k);
    v2f b1 = *(const v2f*)(B1 + (size_t)k * Tn);
    v2f b2 = *(const v2f*)(B2 + (size_t)k * Tn);
    cX = wmma_f32(a, b1, cX);
    cJ = wmma_f32(a, b2, cJ);
  }

  const int t = t0 + row;
  const float tn = ws[OFF_TN + t];
  const float td = ws[OFF_TD + t];
#pragma unroll
  for (int v = 0; v < 8; ++v) {
    const int m = m0 + v + 8 * hi;  // C/D layout: M = v + 8*hi, N = lane&15
    float d2 = ws[OFF_XN + m] - 2.0f * cX[v] + tn;
    float xd = sqrtf(fmaxf(d2, 0.0f));
    float ex = PREF * expf(-xd);
    float dt = cJ[v] - td;
    float a1 = ex * dt;            // exp_xs * dot
    float a2 = ex * (1.0f + xd);   // exp_xs * (1 + x_dist)
    size_t o = (size_t)m * Tn + t;
    ws[OFF_W1 + o] = a1;
    ws[OFF_W2 + o] = a2;
    ws[OFF_E  + o] = a2 * dt;      // energy summand
  }
}

// ---------------------------------------------------------------------------
// Kernel 4: F1a = w1 @ qxt (128x224), F2 = w2 @ Jx.  K = T = 4096.
// 112 tiles = 14 blocks x 8 waves.
// ---------------------------------------------------------------------------
__global__ void gemm2(float* __restrict__ ws) {
  const int wave = threadIdx.x >> 5;
  const int lane = threadIdx.x & 31;
  const int tile = blockIdx.x * 8 + wave;  // 0..111
  const int mt = tile & 7;
  const int dt = tile >> 3;                // 0..13
  const int m0 = mt * 16, d0 = dt * 16;
  const int row = lane & 15, hi = lane >> 4;

  const float* A1 = ws + OFF_W1 + (size_t)(m0 + row) * Tn + 2 * hi;
  const float* A2 = ws + OFF_W2 + (size_t)(m0 + row) * Tn + 2 * hi;
  const float* B1 = ws + OFF_BT2 + (size_t)2 * hi * DP + 2 * (d0 + row);
  const float* B2 = ws + OFF_BJ2 + (size_t)2 * hi * DP + 2 * (d0 + row);

  v8f c1 = {};
  v8f c2 = {};
#pragma unroll 4
  for (int k = 0; k < Tn; k += 4) {
    if ((k & 63) == 0 && k + 256 < Tn) {   // uniform branch, EXEC untouched
      __builtin_prefetch(A1 + k + 256, 0, 0);
      __builtin_prefetch(B1 + (size_t)(k + 256) * DP, 0, 0);
    }
    v2f a1 = *(const v2f*)(A1 + k);
    v2f a2 = *(const v2f*)(A2 + k);
    v2f b1 = *(const v2f*)(B1 + (size_t)k * DP);
    v2f b2 = *(const v2f*)(B2 + (size_t)k * DP);
    c1 = wmma_f32(a1, b1, c1);
    c2 = wmma_f32(a2, b2, c2);
  }
#pragma unroll
  for (int v = 0; v < 8; ++v) {
    const int m = m0 + v + 8 * hi;
    ws[OFF_F1A + (size_t)m * DP + d0 + row] = c1[v];
    ws[OFF_F2  + (size_t)m * DP + d0 + row] = c2[v];
  }
}

// ---------------------------------------------------------------------------
// Kernel 5: row sums S1[m] = sum_t w1, Esum[m] = sum_t e.
// ---------------------------------------------------------------------------
__global__ void rowsum(float* __restrict__ ws) {
  __shared__ float sA[256], sB[256];
  const int m = blockIdx.x, tid = threadIdx.x;
  float a = 0.0f, b = 0.0f;
  for (int t = tid; t < Tn; t += 256) {
    a += ws[OFF_W1 + (size_t)m * Tn + t];
    b += ws[OFF_E  + (size_t)m * Tn + t];
  }
  sA[tid] = a; sB[tid] = b;
  __syncthreads();
  for (int s = 128; s > 0; s >>= 1) {
    if (tid < s) { sA[tid] += sA[tid + s]; sB[tid] += sB[tid + s]; }
    __syncthreads();
  }
  if (tid == 0) { ws[OFF_S1 + m] = sA[0]; ws[OFF_ES + m] = sB[0]; }
}

// ---------------------------------------------------------------------------
// Kernel 6: coef[m,d] = ((qxs*S1 - F1a) - F2) * std / r^3
// ---------------------------------------------------------------------------
__global__ void coefk(float* __restrict__ ws) {
  const int idx = blockIdx.x * 256 + threadIdx.x;
  if (idx >= Mn * Dn) return;
  const int m = idx / Dn, d = idx % Dn;
  float F1  = ws[OFF_QXSA + (size_t)m * KP + d] * ws[OFF_S1 + m]
            - ws[OFF_F1A + (size_t)m * DP + d];
  float Fsx = (F1 - ws[OFF_F2 + (size_t)m * DP + d]) * STDf;
  float r   = ws[OFF_R + (size_t)m * Dn + d];
  ws[OFF_COEF + (size_t)m * Dn + d] = Fsx / (r * r * r);
}

// ---------------------------------------------------------------------------
// Kernel 7: Fs[m,b,c] = sum_a full[a,b] * (Rs[m,a,c] - Rs[m,b,c])
// ---------------------------------------------------------------------------
__global__ void assemble(const float* __restrict__ Rs,
                         const float* __restrict__ ws,
                         float* __restrict__ out) {
  __shared__ float rs[NA * 3];
  __shared__ float cf[Dn];
  const int m = blockIdx.x, tid = threadIdx.x;  // 64 threads
  if (tid < NA * 3) rs[tid] = Rs[(size_t)m * NA * 3 + tid];
  for (int i = tid; i < Dn; i += 64) cf[i] = ws[OFF_COEF + (size_t)m * Dn + i];
  __syncthreads();
  if (tid < NA * 3) {
    const int b = tid / 3, cc = tid % 3;
    const float rb = rs[b * 3 + cc];
    float acc = 0.0f;
    for (int a = 0; a < NA; ++a) {
      if (a == b) continue;
      const int ii = a > b ? a : b;
      const int jj = a > b ? b : a;
      acc += cf[ii * (ii - 1) / 2 + jj] * (rs[a * 3 + cc] - rb);
    }
    out[Mn + (size_t)m * NA * 3 + tid] = acc;
  }
}

// ---------------------------------------------------------------------------
// Kernel 8: Es[m] = C + std * Esum[m] / q
// ---------------------------------------------------------------------------
__global__ void esk(const float* __restrict__ ws, float* __restrict__ out) {
  const int m = threadIdx.x;
  out[m] = Cf + STDf * ws[OFF_ES + m] / Qf;
}

// ---------------------------------------------------------------------------
extern "C" void kernel_launch(void* const* d_in, const int* in_sizes, int n_in,
                              void* d_out, int out_size, void* d_ws, size_t ws_size,
                              hipStream_t stream) {
  (void)in_sizes; (void)n_in; (void)out_size; (void)ws_size;
  const float* Rs       = (const float*)d_in[0];
  const float* xs_train = (const float*)d_in[1];
  const float* Jx       = (const float*)d_in[2];
  float* out = (float*)d_out;
  float* ws  = (float*)d_ws;

  prep_mol  <<<Mn, 256, 0, stream>>>(Rs, ws);
  prep_train<<<Tn, 256, 0, stream>>>(xs_train, Jx, ws);
  gemm1     <<<256, 256, 0, stream>>>(ws);          // 2048 wave-tiles
  gemm2     <<<14, 256, 0, stream>>>(ws);           // 112 wave-tiles
  rowsum    <<<Mn, 256, 0, stream>>>(ws);
  coefk     <<<(Mn * Dn + 255) / 256, 256, 0, stream>>>(ws);
  assemble  <<<Mn, 64, 0, stream>>>(Rs, ws, out);
  esk       <<<1, Mn, 0, stream>>>(ws, out);
}